// Detect_27960237097706
// MI455X (gfx1250) — compile-verified
//
#include <hip/hip_runtime.h>
#include <hip/hip_bf16.h>
#include <stdint.h>

// ---------------------------------------------------------------------------
// RefineDet Detect post-processing for MI455X (gfx1250, wave32, WGP, WMMA).
//
// Pipeline (all on `stream`, serialized by stream order):
//   K0: zero d_out (background class plane + padding slots must be 0).
//   K1: decode boxes + softmax/mask scores            (bandwidth bound, ~12MB)
//   K2: per-(b,c) exact sorted top-1000 via in-LDS bitonic sort of packed
//       u64 keys (256KB dynamic LDS -- fits CDNA5's 320KB WGP LDS).
//   K3: per-(b,c) NMS: async-DMA gather of top boxes into LDS, WMMA-assisted
//       pairwise-IoU suppression bitmask (128KB in LDS), wave32 serial greedy
//       scan, stable compaction.
// ---------------------------------------------------------------------------

#define NUM_CLASSES 5
#define TOPK        1000
#define NMS_THRESH  0.45f
#define CONF_THRESH 0.01f
#define OBJ_THRESH  0.01f
#define VAR0        0.1f
#define VAR1        0.2f
#define BB          8
#define PP          24564

typedef float  v2f __attribute__((ext_vector_type(2)));
typedef float  v8f __attribute__((ext_vector_type(8)));

// ---------------------------------------------------------------------- K0
__global__ void k0_zero(float* out, int n) {
    int i = blockIdx.x * blockDim.x + threadIdx.x;
    if (i < n) out[i] = 0.0f;
}

// ---------------------------------------------------------------------- K1
// decode(odm_loc, center_size(decode(arm_loc, prior))) collapses:
// center_size(decode(l,p)) == (xy, wh) of the first decode directly.
__global__ void k1_decode(const float* __restrict__ arm_loc,
                          const float* __restrict__ arm_conf,
                          const float* __restrict__ odm_loc,
                          const float* __restrict__ odm_conf,
                          const float* __restrict__ prior,
                          float* __restrict__ boxes,     // [B*P][4]
                          float* __restrict__ scores)    // [B*4][P]
{
    int gid = blockIdx.x * blockDim.x + threadIdx.x;
    if (gid >= BB * PP) return;
    int b = gid / PP, p = gid - b * PP;

    const float* oc = odm_conf + (size_t)gid * NUM_CLASSES;
    __builtin_prefetch(oc, 0, 0);  // global_prefetch_b8 (strided 20B rows)

    float4 pr = ((const float4*)prior)[p];
    float4 al = ((const float4*)arm_loc)[gid];
    float4 ol = ((const float4*)odm_loc)[gid];
    float2 ac = ((const float2*)arm_conf)[gid];

    // arm softmax[...,1] = sigmoid(c1 - c0); object mask
    float obj  = 1.0f / (1.0f + __expf(ac.x - ac.y));
    float objm = (obj > OBJ_THRESH) ? 1.0f : 0.0f;

    // odm softmax over 5 classes (max-subtracted)
    float c0 = oc[0], c1 = oc[1], c2 = oc[2], c3 = oc[3], c4 = oc[4];
    float mx = fmaxf(fmaxf(fmaxf(c0, c1), fmaxf(c2, c3)), c4);
    float e0 = __expf(c0 - mx), e1 = __expf(c1 - mx), e2 = __expf(c2 - mx),
          e3 = __expf(c3 - mx), e4 = __expf(c4 - mx);
    float inv = 1.0f / (e0 + e1 + e2 + e3 + e4);
    float s1 = e1 * inv * objm, s2 = e2 * inv * objm,
          s3 = e3 * inv * objm, s4 = e4 * inv * objm;
    const float NEG = -__builtin_inff();
    scores[((size_t)(b * 4 + 0)) * PP + p] = (s1 > CONF_THRESH) ? s1 : NEG;
    scores[((size_t)(b * 4 + 1)) * PP + p] = (s2 > CONF_THRESH) ? s2 : NEG;
    scores[((size_t)(b * 4 + 2)) * PP + p] = (s3 > CONF_THRESH) ? s3 : NEG;
    scores[((size_t)(b * 4 + 3)) * PP + p] = (s4 > CONF_THRESH) ? s4 : NEG;

    // arm decode -> (cx,cy,w,h)
    float cx = pr.x + al.x * VAR0 * pr.z;
    float cy = pr.y + al.y * VAR0 * pr.w;
    float w1 = pr.z * __expf(al.z * VAR1);
    float h1 = pr.w * __expf(al.w * VAR1);
    // odm decode on top
    float cx2 = cx + ol.x * VAR0 * w1;
    float cy2 = cy + ol.y * VAR0 * h1;
    float w2  = w1 * __expf(ol.z * VAR1);
    float h2  = h1 * __expf(ol.w * VAR1);
    float4 bx;
    bx.x = cx2 - 0.5f * w2;  bx.y = cy2 - 0.5f * h2;
    bx.z = cx2 + 0.5f * w2;  bx.w = cy2 + 0.5f * h2;
    ((float4*)boxes)[gid] = bx;
}

// ---------------------------------------------------------------------- K2
// Sorted top-K per (b,c): full bitonic sort of 32768 u64 keys in 256KB LDS.
// key = (~monotone(score)) << 32 | idx  -> ascending sort == descending
// score with ascending-index ties, exactly matching lax.top_k.
__global__ void __launch_bounds__(1024, 1)
k2_topk(const float* __restrict__ scores,
        float* __restrict__ topS, unsigned* __restrict__ topI)
{
    extern __shared__ char dynsmem[];
    unsigned long long* keys = (unsigned long long*)dynsmem;  // [32768]
    const int N = 32768;
    int t = threadIdx.x;
    const float* base = scores + (size_t)blockIdx.x * PP;

    for (int i = t; i < N; i += 1024) {
        unsigned long long k;
        if (i < PP) {
            unsigned u = __float_as_uint(base[i]);
            unsigned m = (u & 0x80000000u) ? ~u : (u | 0x80000000u);
            k = ((unsigned long long)(~m) << 32) | (unsigned)i;
        } else {
            k = ~0ULL;  // padding sorts last
        }
        keys[i] = k;
    }
    __syncthreads();

    for (unsigned kk = 2; kk <= (unsigned)N; kk <<= 1) {
        for (unsigned j = kk >> 1; j > 0; j >>= 1) {
            for (int i = t; i < N; i += 1024) {
                unsigned p = (unsigned)i ^ j;
                if (p > (unsigned)i) {
                    unsigned long long a = keys[i], b = keys[p];
                    bool asc = (((unsigned)i & kk) == 0);
                    if ((a > b) == asc) { keys[i] = b; keys[p] = a; }
                }
            }
            __syncthreads();
        }
    }

    if (t < TOPK) {
        unsigned long long k = keys[t];
        unsigned m    = ~(unsigned)(k >> 32);
        unsigned bits = (m & 0x80000000u) ? (m ^ 0x80000000u) : ~m;
        topS[(size_t)blockIdx.x * TOPK + t] = __uint_as_float(bits);
        topI[(size_t)blockIdx.x * TOPK + t] = (unsigned)k;
    }
}

// ---------------------------------------------------------------------- K3
// NMS per (b,c). Dynamic LDS layout (155,904 B):
//   sBoxes [1024] float4   (16384 B)   gathered via async DMA
//   sArea  [1024] float    ( 4096 B)
//   sScore [1024] float    ( 4096 B)
//   sMask  [1024*32] u32   (131072 B)  suppression bitmask rows
//   sKeep  [32] u32, sPref [32] u32
__global__ void __launch_bounds__(1024, 1)
k3_nms(const float* __restrict__ boxes,
       const float* __restrict__ topS, const unsigned* __restrict__ topI,
       float* __restrict__ out)
{
    extern __shared__ char dynsmem[];
    float4*   sBoxes = (float4*)dynsmem;
    float*    sArea  = (float*)(dynsmem + 16384);
    float*    sScore = (float*)(dynsmem + 16384 + 4096);
    unsigned* sMask  = (unsigned*)(dynsmem + 24576);
    unsigned* sKeep  = (unsigned*)(dynsmem + 24576 + 131072);
    unsigned* sPref  = sKeep + 32;

    const int t = threadIdx.x, lane = t & 31, wave = t >> 5;
    const int bc = blockIdx.x;           // b*4 + (class-1)
    const int b  = bc >> 2, cls = bc & 3;

    // ---- Stage A: gather top boxes via async global->LDS DMA --------------
    float myScore = -__builtin_inff();
    if (t < TOPK) {
        myScore = topS[(size_t)bc * TOPK + t];
        unsigned idx = topI[(size_t)bc * TOPK + t];
        const float* src = boxes + ((size_t)b * PP + idx) * 4;
        unsigned long long ga = (unsigned long long)(uintptr_t)src;
        unsigned ldsOff = (unsigned)(uintptr_t)(void*)&sBoxes[t];
        asm volatile("global_load_async_to_lds_b128 %0, %1, off"
                     :: "v"(ldsOff), "v"(ga) : "memory");
    } else {
        sBoxes[t] = make_float4(0.f, 0.f, 0.f, 0.f);   // pad rows
    }
    for (int i = t; i < 1024 * 32; i += 1024) sMask[i] = 0u;
    if (t < 32) sKeep[t] = 0u;
    asm volatile("s_wait_asynccnt 0x0" ::: "memory");
    __syncthreads();

    float4 mybx = sBoxes[t];
    sArea[t]  = (mybx.z - mybx.x) * (mybx.w - mybx.y);
    sScore[t] = myScore;
    if (t < TOPK && myScore > -3.0e38f)
        atomicOr(&sKeep[t >> 5], 1u << (t & 31));
    __syncthreads();

    // ---- Stage B: suppression bitmask, WMMA-assisted union sums ----------
    // Per 16x16 tile: union(m,n) = area[m]+area[n] via one
    // v_wmma_f32_16x16x4_f32 rank-2 outer-sum; intersection on VALU.
    const int half = lane >> 4, col = lane & 15;
    for (int tile = wave; tile < 64 * 64; tile += 32) {
        int tr = tile >> 6, tc = tile & 63;
        if (tc < tr) continue;           // whole tile has n <= m
        v2f av, bv;
        av.x = half ? 0.0f : sArea[tr * 16 + col];   // A: K0=area_row, K1=1
        av.y = half ? 0.0f : 1.0f;
        bv.x = half ? 0.0f : 1.0f;                   // B: K0=1, K1=area_col
        bv.y = half ? 0.0f : sArea[tc * 16 + col];
        v8f c = {};
        c = __builtin_amdgcn_wmma_f32_16x16x4_f32(
                false, av, false, bv, (short)0, c, false, false);

        float4 bn = sBoxes[tc * 16 + col];           // column box per lane
        int mbase = tr * 16 + half * 8;
        int n = tc * 16 + col;
        #pragma unroll
        for (int r = 0; r < 8; ++r) {
            int m = mbase + r;
            float4 bm = sBoxes[m];
            float w = fminf(bm.z, bn.z) - fmaxf(bm.x, bn.x);
            float h = fminf(bm.w, bn.w) - fmaxf(bm.y, bn.y);
            w = fmaxf(w, 0.0f); h = fmaxf(h, 0.0f);
            float inter = w * h;
            float uni   = c[r] - inter;              // area_m + area_n - inter
            bool pred = (inter > NMS_THRESH * uni) && (n > m) &&
                        (n < TOPK) && (m < TOPK);
            unsigned bal = (unsigned)__ballot(pred); // wave32: 32 bits
            if (lane == 0) {
                unsigned lo = bal & 0xFFFFu, hi = bal >> 16;
                int jw = tc >> 1, sh = (tc & 1) * 16;
                if (lo) atomicOr(&sMask[(tr * 16 + r) * 32 + jw], lo << sh);
                if (hi) atomicOr(&sMask[(tr * 16 + 8 + r) * 32 + jw], hi << sh);
            }
        }
    }
    __syncthreads();

    // ---- Stage C: greedy serial scan, one wave32 (1024 bits = 32 lanes) ---
    if (wave == 0) {
        unsigned kw = sKeep[lane];
        for (int i = 0; i < TOPK; ++i) {
            unsigned cur = (unsigned)__shfl((int)kw, i >> 5, 32);
            if ((cur >> (i & 31)) & 1u)
                kw &= ~sMask[i * 32 + lane];
        }
        sKeep[lane] = kw;
    }
    __syncthreads();

    if (t == 0) {
        unsigned acc = 0;
        for (int w = 0; w < 32; ++w) { sPref[w] = acc; acc += __popc(sKeep[w]); }
    }
    __syncthreads();

    // ---- Stage D: stable compaction into out[b][cls+1][pos][0..4] ---------
    if (t < TOPK) {
        unsigned word = sKeep[t >> 5], bpos = (unsigned)(t & 31);
        if ((word >> bpos) & 1u) {
            unsigned pos = sPref[t >> 5] + __popc(word & ((1u << bpos) - 1u));
            float4 bx = sBoxes[t];
            size_t base = ((((size_t)b * NUM_CLASSES + (cls + 1)) * TOPK) + pos) * 5;
            out[base + 0] = sScore[t];
            out[base + 1] = bx.x; out[base + 2] = bx.y;
            out[base + 3] = bx.z; out[base + 4] = bx.w;
        }
    }
}

// ---------------------------------------------------------------------------
extern "C" void kernel_launch(void* const* d_in, const int* in_sizes, int n_in,
                              void* d_out, int out_size, void* d_ws, size_t ws_size,
                              hipStream_t stream) {
    const float* arm_loc  = (const float*)d_in[0];
    const float* arm_conf = (const float*)d_in[1];
    const float* odm_loc  = (const float*)d_in[2];
    const float* odm_conf = (const float*)d_in[3];
    const float* prior    = (const float*)d_in[4];
    float* out = (float*)d_out;

    // Workspace carve (~6.3 MB total)
    float*    boxes  = (float*)d_ws;                       // B*P*4
    float*    scores = boxes  + (size_t)BB * PP * 4;       // B*4*P
    float*    topS   = scores + (size_t)BB * 4 * PP;       // 32*K
    unsigned* topI   = (unsigned*)(topS + 32 * TOPK);      // 32*K

    k0_zero<<<(out_size + 255) / 256, 256, 0, stream>>>(out, out_size);

    int nthreads = BB * PP;
    k1_decode<<<(nthreads + 255) / 256, 256, 0, stream>>>(
        arm_loc, arm_conf, odm_loc, odm_conf, prior, boxes, scores);

    k2_topk<<<32, 1024, 32768 * sizeof(unsigned long long), stream>>>(
        scores, topS, topI);

    k3_nms<<<32, 1024, 155904, stream>>>(boxes, topS, topI, out);
}